// AutoencoderInverseAffine_48249662603934
// MI455X (gfx1250) — compile-verified
//
#include <hip/hip_runtime.h>

typedef float v4f __attribute__((ext_vector_type(4)));
typedef int   vi4 __attribute__((vector_size(4 * sizeof(int))));

#define AS1 __attribute__((address_space(1)))
#define AS3 __attribute__((address_space(3)))

// Problem constants from the reference (N_DIM=8, N_COMP=4, NX_UNIQUE=16)
constexpr int NDIM = 8;
constexpr int PSI  = 32;   // N_COMP * N_DIM
constexpr int TBL  = 512;  // NX_UNIQUE * N_COMP * N_DIM

__global__ __launch_bounds__(256) void inv_affine_kernel(
    const float* __restrict__ samples,
    const float* __restrict__ mus_orig,
    const float* __restrict__ mus,
    const float* __restrict__ psi,
    const int*   __restrict__ idx_symb,
    const int*   __restrict__ idx_comp,
    float*       __restrict__ out,
    int n_samp)
{
    __shared__ __align__(16) float s_mus [TBL];
    __shared__ __align__(16) float s_musO[TBL];
    __shared__ __align__(16) float s_psi [PSI];
    __shared__ __align__(16) float s_scale[PSI];
    __shared__ __align__(16) float s_bias [TBL];

    const int tid = threadIdx.x;

    // ---- Stage the (tiny) parameter tables into LDS -------------------------
#if __has_builtin(__builtin_amdgcn_global_load_async_to_lds_b128)
    // CDNA5 async global->LDS copies (ASYNCcnt path). 16B per active lane.
    if (tid < TBL / 4) {
        __builtin_amdgcn_global_load_async_to_lds_b128(
            (AS1 vi4*)(mus + tid * 4),      (AS3 vi4*)(s_mus  + tid * 4), 0, 0);
        __builtin_amdgcn_global_load_async_to_lds_b128(
            (AS1 vi4*)(mus_orig + tid * 4), (AS3 vi4*)(s_musO + tid * 4), 0, 0);
    }
    if (tid < PSI / 4) {
        __builtin_amdgcn_global_load_async_to_lds_b128(
            (AS1 vi4*)(psi + tid * 4),      (AS3 vi4*)(s_psi  + tid * 4), 0, 0);
    }
  #if __has_builtin(__builtin_amdgcn_s_wait_asynccnt)
    __builtin_amdgcn_s_wait_asynccnt(0);
  #else
    asm volatile("s_wait_asynccnt 0" ::: "memory");
  #endif
    __syncthreads();
#else
    for (int i = tid; i < TBL; i += 256) { s_mus[i] = mus[i]; s_musO[i] = mus_orig[i]; }
    if (tid < PSI) s_psi[tid] = psi[tid];
    __syncthreads();
#endif

    // ---- Fuse the affine math once per block --------------------------------
    // out = (s - mu)/psi + mu_orig  ==  s*scale + bias
    // scale = 1/psi ; bias = mu_orig - mu*scale
    if (tid < PSI) s_scale[tid] = 1.0f / s_psi[tid];
    __syncthreads();
    for (int i = tid; i < TBL; i += 256)
        s_bias[i] = fmaf(-s_mus[i], s_scale[i & (PSI - 1)], s_musO[i]);
    __syncthreads();

    // ---- Streaming main loop: one float4 (half a sample) per thread --------
    const v4f* __restrict__ s4 = (const v4f*)samples;
    v4f*       __restrict__ o4 = (v4f*)out;

    const int nvec   = n_samp * 2;                 // number of float4 chunks
    const int stride = gridDim.x * blockDim.x;

    for (int j = blockIdx.x * blockDim.x + tid; j < nvec; j += stride) {
        const int n    = j >> 1;
        const int half = (j & 1) << 2;             // 0 or 4 floats within the sample
        const int symb = idx_symb[n];
        const int comp = idx_comp[n];

        // 512 MB of stream traffic, touched exactly once -> non-temporal b128.
        v4f a = __builtin_nontemporal_load(s4 + j);

        const v4f sc = *(const v4f*)(s_scale + comp * NDIM + half);
        const v4f bi = *(const v4f*)(s_bias + symb * PSI + comp * NDIM + half);

        v4f r = a * sc + bi;                       // contracts to packed FMAs

        __builtin_nontemporal_store(r, o4 + j);
    }
}

extern "C" void kernel_launch(void* const* d_in, const int* in_sizes, int n_in,
                              void* d_out, int out_size, void* d_ws, size_t ws_size,
                              hipStream_t stream) {
    // Input order per setup_inputs():
    // 0 samples_ (f32), 1 mus_orig_ (f32), 2 mus_ (f32), 3 psi_c_ (f32),
    // 4 idx_symb_ (i32), 5 idx_comp_ (i32), 6 n_samp_ (scalar), 7 n_dim_ (scalar)
    const float* samples  = (const float*)d_in[0];
    const float* mus_orig = (const float*)d_in[1];
    const float* mus      = (const float*)d_in[2];
    const float* psi      = (const float*)d_in[3];
    const int*   idx_symb = (const int*)d_in[4];
    const int*   idx_comp = (const int*)d_in[5];
    float* out = (float*)d_out;

    const int n_samp = in_sizes[4];                // == N_SAMP
    const int nvec   = n_samp * 2;

    const int threads = 256;                       // 8 waves/WGP-slot under wave32
    int blocks = (nvec + threads - 1) / threads;
    if (blocks > 16384) blocks = 16384;            // grid-stride covers the rest

    hipLaunchKernelGGL(inv_affine_kernel, dim3(blocks), dim3(threads), 0, stream,
                       samples, mus_orig, mus, psi, idx_symb, idx_comp, out, n_samp);
}